// EMAFilter_9680856285442
// MI455X (gfx1250) — compile-verified
//
#include <hip/hip_runtime.h>

// EMA filter as batched WMMA GEMM on gfx1250 (MI455X, wave32).
//
// out[d, 16a+b] = sum_n (c[d,n] * (q^16)^a) * (q^b),  c = p*gamma
//   => per-d GEMM:  Out[256x16] = U[256x16] x V[16x16]
// computed with chained V_WMMA_F32_16X16X4_F32 (full f32 precision).
//
// Each wave owns 4 consecutive 16-row a-tiles of one d:
//  - B (q^b, 16x16) is built once and reused for all 4 tiles
//  - A advances between tiles by a columnwise multiply with q^256
//    (A_{t+1}[m,k] = A_t[m,k] * q_k^256), avoiding fresh exps per tile.

typedef __attribute__((ext_vector_type(2))) float v2f;
typedef __attribute__((ext_vector_type(8))) float v8f;

#define D_DIM 2048
#define N_DIM 16
#define L_DIM 4096
#define TILES_PER_D 16          // a in [0,256), 16 rows per tile
#define TILES_PER_WAVE 4
#define WAVES_TOTAL (D_DIM * (TILES_PER_D / TILES_PER_WAVE))   // 8192
#define BLOCK_THREADS 256
#define GRID_BLOCKS (WAVES_TOTAL / (BLOCK_THREADS / 32))       // 1024

__global__ __launch_bounds__(BLOCK_THREADS)
void EMAFilter_wmma_kernel(const float* __restrict__ p,
                           const float* __restrict__ q,
                           const float* __restrict__ gamma,
                           float* __restrict__ out)
{
    const int wave = (int)((blockIdx.x * (unsigned)BLOCK_THREADS + threadIdx.x) >> 5);
    const int lane = (int)(threadIdx.x & 31u);
    const int d    = wave >> 2;                 // 0..2047
    const int t0   = (wave & 3) * TILES_PER_WAVE; // first a-tile of this wave
    const int half = lane >> 4;                 // K pair {0,1} vs {2,3} (A & B), row+8 (C/D)
    const int m    = lane & 15;                 // M for A, N for B/C/D

    const int base = d * N_DIM;

    // A[m=a_local, k]: U[a, n] = c_n * q_n^(16a), a = 16*t + m
    const float a_pow0 = (float)((16 * t0 + m) * 16);
    // B[k, n=b]:       V[n, b] = q_n^b, b = m
    const float b_pow = (float)m;

    // Per-lane: this lane touches n = 4*kk + 2*half (+1) for kk = 0..3.
    v2f Amat[4], Bmat[4], Rmat[4];
#pragma unroll
    for (int kk = 0; kk < 4; ++kk) {
        const int na = 4 * kk + 2 * half;
        const int nb = na + 1;
        const float lqx = logf(q[base + na]);
        const float lqy = logf(q[base + nb]);
        const float ccx = p[base + na] * gamma[base + na];
        const float ccy = p[base + nb] * gamma[base + nb];
        Bmat[kk].x = __expf(b_pow * lqx);
        Bmat[kk].y = __expf(b_pow * lqy);
        Amat[kk].x = ccx * __expf(a_pow0 * lqx);
        Amat[kk].y = ccy * __expf(a_pow0 * lqy);
        Rmat[kk].x = __expf(256.0f * lqx);      // q^256: a-step of 16 rows
        Rmat[kk].y = __expf(256.0f * lqy);
    }

    // C/D layout: VGPR r holds row M = r + 8*half, col N = m.
    // l = 16*a + b = 256*t + 128*half + 16*r + m
    float* o = out + (size_t)d * L_DIM + 256 * t0 + 128 * half + m;

#pragma unroll
    for (int tt = 0; tt < TILES_PER_WAVE; ++tt) {
        v8f acc = {};
#pragma unroll
        for (int kk = 0; kk < 4; ++kk) {
            // 8 args: (neg_a, A, neg_b, B, c_mod, C, reuse_a, reuse_b)
            acc = __builtin_amdgcn_wmma_f32_16x16x4_f32(
                false, Amat[kk], false, Bmat[kk], (short)0, acc, false, false);
        }
#pragma unroll
        for (int r = 0; r < 8; ++r)
            o[16 * r] = acc[r];
        o += 256;

        if (tt < TILES_PER_WAVE - 1) {
#pragma unroll
            for (int kk = 0; kk < 4; ++kk) {
                Amat[kk].x *= Rmat[kk].x;
                Amat[kk].y *= Rmat[kk].y;
            }
        }
    }
}

extern "C" void kernel_launch(void* const* d_in, const int* in_sizes, int n_in,
                              void* d_out, int out_size, void* d_ws, size_t ws_size,
                              hipStream_t stream)
{
    const float* p     = (const float*)d_in[0];   // (D, N, 1)
    const float* q     = (const float*)d_in[1];   // (D, N, 1)
    const float* gamma = (const float*)d_in[2];   // (D, N)
    float*       out   = (float*)d_out;           // (D, L)
    (void)in_sizes; (void)n_in; (void)out_size; (void)d_ws; (void)ws_size;

    EMAFilter_wmma_kernel<<<GRID_BLOCKS, BLOCK_THREADS, 0, stream>>>(p, q, gamma, out);
}